// GDN_51453708206596
// MI455X (gfx1250) — compile-verified
//
#include <hip/hip_runtime.h>
#include <math.h>

#define Bt 32
#define Nn 1024
#define Ff 10
#define Dd 64
#define Mm 4
#define Rr 8
#define Kk 20
#define BNTOT (Bt * Nn)              // 32768
#define BIGSZ ((size_t)BNTOT * Dd)   // 2097152 floats

typedef __attribute__((ext_vector_type(2))) float v2f;
typedef __attribute__((ext_vector_type(4))) float v4f;
typedef __attribute__((ext_vector_type(8))) float v8f;

// Branchless nan_to_num(nan=0, posinf=1e4, neginf=-1e4): lowers to v_cndmask,
// no EXEC-mask branches near the WMMA region.
__device__ __forceinline__ float nanfix(float x) {
  float y = (x != x) ? 0.f : x;
  y = (y == INFINITY) ? 1e4f : y;
  y = (y == -INFINITY) ? -1e4f : y;
  return y;
}
__device__ __forceinline__ float lrelu(float x) { return x > 0.f ? x : 0.2f * x; }

// ---------------------------------------------------------------------------
// K0: per-node prep: h_it = data@W_cond+b ; e_it = leaky(h@W_ap+b)@W_av ;
//     x_lin = data@W_gnn ; s_i/s_j partial = x_lin . att_{i,j}
// ---------------------------------------------------------------------------
__global__ __launch_bounds__(256) void k_node_prep(
    const float* __restrict__ data, const float* __restrict__ W_cond,
    const float* __restrict__ b_cond, const float* __restrict__ W_ap,
    const float* __restrict__ b_ap, const float* __restrict__ W_av,
    const float* __restrict__ W_gnn, const float* __restrict__ att_i,
    const float* __restrict__ att_j, float* __restrict__ h_it,
    float* __restrict__ e_it, float* __restrict__ x_lin,
    float* __restrict__ s_i, float* __restrict__ s_j) {
  int idx = blockIdx.x * 256 + threadIdx.x;  // [0, 32768)
  const float* x = data + (size_t)idx * Ff;
  float xr[Ff];
#pragma unroll
  for (int f = 0; f < Ff; ++f) xr[f] = x[f];

  float h[Dd];
#pragma unroll 8
  for (int d = 0; d < Dd; ++d) {
    float a = b_cond[d];
#pragma unroll
    for (int f = 0; f < Ff; ++f) a += xr[f] * W_cond[f * Dd + d];
    h[d] = a;
  }
#pragma unroll
  for (int d4 = 0; d4 < Dd / 4; ++d4)
    *(v4f*)(h_it + (size_t)idx * Dd + 4 * d4) =
        (v4f){h[4 * d4], h[4 * d4 + 1], h[4 * d4 + 2], h[4 * d4 + 3]};

  float e = 0.f;
#pragma unroll 4
  for (int d2 = 0; d2 < Dd; ++d2) {
    float a = b_ap[d2];
#pragma unroll 8
    for (int d = 0; d < Dd; ++d) a += h[d] * W_ap[d * Dd + d2];
    e += lrelu(a) * W_av[d2];
  }
  e_it[idx] = nanfix(e);

  float si = 0.f, sj = 0.f;
  float xl[Dd];
#pragma unroll 8
  for (int d = 0; d < Dd; ++d) {
    float v = 0.f;
#pragma unroll
    for (int f = 0; f < Ff; ++f) v += xr[f] * W_gnn[f * Dd + d];
    xl[d] = v;
    si += v * att_i[d];
    sj += v * att_j[d];
  }
#pragma unroll
  for (int d4 = 0; d4 < Dd / 4; ++d4)
    *(v4f*)(x_lin + (size_t)idx * Dd + 4 * d4) =
        (v4f){xl[4 * d4], xl[4 * d4 + 1], xl[4 * d4 + 2], xl[4 * d4 + 3]};
  s_i[idx] = si;
  s_j[idx] = sj;
}

// ---------------------------------------------------------------------------
// K1: per-batch attention pooling (beta softmax, h_sys) + gumbel top-2 routing
// ---------------------------------------------------------------------------
__global__ __launch_bounds__(256) void k_pool_route(
    const float* __restrict__ h_it, const float* __restrict__ e_it,
    const float* __restrict__ W_r, const float* __restrict__ b_r,
    const float* __restrict__ gumbel, float* __restrict__ out_hsys,
    float* __restrict__ out_pisoft, float* __restrict__ pi_t) {
  __shared__ float red[256];
  __shared__ float beta_sh[Nn];
  __shared__ float hsys_sh[Dd];
  int b = blockIdx.x, tid = threadIdx.x;
  const float* e = e_it + (size_t)b * Nn;

  float m = -INFINITY;
  for (int n = tid; n < Nn; n += 256) m = fmaxf(m, e[n]);
  red[tid] = m;
  __syncthreads();
  for (int s = 128; s > 0; s >>= 1) {
    if (tid < s) red[tid] = fmaxf(red[tid], red[tid + s]);
    __syncthreads();
  }
  float mx = red[0];
  __syncthreads();
  float ssum = 0.f;
  for (int n = tid; n < Nn; n += 256) {
    float ev = expf(e[n] - mx);
    beta_sh[n] = ev;
    ssum += ev;
  }
  red[tid] = ssum;
  __syncthreads();
  for (int s = 128; s > 0; s >>= 1) {
    if (tid < s) red[tid] += red[tid + s];
    __syncthreads();
  }
  float tot = red[0];
  __syncthreads();

  if (tid < Dd) {
    float acc = 0.f;
    const float* hb = h_it + (size_t)b * Nn * Dd + tid;
    for (int n = 0; n < Nn; ++n) acc += beta_sh[n] * hb[(size_t)n * Dd];
    acc /= tot;
    hsys_sh[tid] = acc;
    out_hsys[b * Dd + tid] = acc;
  }
  __syncthreads();

  if (tid == 0) {
    float lg[Mm];
#pragma unroll
    for (int mm = 0; mm < Mm; ++mm) {
      float a = b_r[mm] + gumbel[b * Mm + mm];
      for (int d = 0; d < Dd; ++d) a += hsys_sh[d] * W_r[d * Mm + mm];
      lg[mm] = a;  // TAU == 1
    }
    float m2 = lg[0];
#pragma unroll
    for (int mm = 1; mm < Mm; ++mm) m2 = fmaxf(m2, lg[mm]);
    float sm = 0.f, pi[Mm];
#pragma unroll
    for (int mm = 0; mm < Mm; ++mm) {
      pi[mm] = expf(lg[mm] - m2);
      sm += pi[mm];
    }
#pragma unroll
    for (int mm = 0; mm < Mm; ++mm) {
      pi[mm] /= sm;
      out_pisoft[b * Mm + mm] = pi[mm];
    }
    // top-2 (ties -> lowest index first, matching lax.top_k)
    int i1 = 0;
#pragma unroll
    for (int mm = 1; mm < Mm; ++mm)
      if (pi[mm] > pi[i1]) i1 = mm;
    int i2 = (i1 == 0) ? 1 : 0;
#pragma unroll
    for (int mm = 0; mm < Mm; ++mm)
      if (mm != i1 && pi[mm] > pi[i2]) i2 = mm;
    float s2 = fmaxf(pi[i1] + pi[i2], 1e-12f);
#pragma unroll
    for (int mm = 0; mm < Mm; ++mm) {
      float ph = (mm == i1) ? pi[i1] : (mm == i2 ? pi[i2] : 0.f);
      pi_t[b * Mm + mm] = ph / s2;  // forward value of ST estimator == pi_hard
    }
  }
}

// ---------------------------------------------------------------------------
// K2: mixed[b,n,:] = sum_m pi_t[b,m]*(e_base + lr_u@lr_v) ; s_i/s_j += mixed.att_em
// ---------------------------------------------------------------------------
__global__ __launch_bounds__(256) void k_mixed(
    const float* __restrict__ pi_t, const float* __restrict__ e_base,
    const float* __restrict__ lr_u, const float* __restrict__ lr_v,
    const float* __restrict__ att_em_i, const float* __restrict__ att_em_j,
    float* __restrict__ mixed, float* __restrict__ s_i, float* __restrict__ s_j) {
  int idx = blockIdx.x * 256 + threadIdx.x;
  int b = idx >> 10, n = idx & (Nn - 1);
  float pv[Mm], ps = 0.f;
#pragma unroll
  for (int mm = 0; mm < Mm; ++mm) {
    pv[mm] = pi_t[b * Mm + mm];
    ps += pv[mm];
  }
  float mix[Dd];
#pragma unroll 8
  for (int d = 0; d < Dd; ++d) mix[d] = ps * e_base[(size_t)n * Dd + d];
#pragma unroll
  for (int mm = 0; mm < Mm; ++mm) {
    float u[Rr];
#pragma unroll
    for (int r = 0; r < Rr; ++r) u[r] = lr_u[((size_t)mm * Nn + n) * Rr + r];
#pragma unroll 8
    for (int d = 0; d < Dd; ++d) {
      float t = 0.f;
#pragma unroll
      for (int r = 0; r < Rr; ++r) t += u[r] * lr_v[((size_t)mm * Rr + r) * Dd + d];
      mix[d] += pv[mm] * t;
    }
  }
  float si = 0.f, sj = 0.f;
#pragma unroll 8
  for (int d = 0; d < Dd; ++d) {
    float v = nanfix(mix[d]);
    mix[d] = v;
    si += v * att_em_i[d];
    sj += v * att_em_j[d];
  }
#pragma unroll
  for (int d4 = 0; d4 < Dd / 4; ++d4)
    *(v4f*)(mixed + (size_t)idx * Dd + 4 * d4) =
        (v4f){mix[4 * d4], mix[4 * d4 + 1], mix[4 * d4 + 2], mix[4 * d4 + 3]};
  s_i[idx] += si;
  s_j[idx] += sj;
}

// ---------------------------------------------------------------------------
// K3 (the WMMA kernel): per (b, 16-row strip):
//   scores strip (16x1024) via v_wmma_f32_16x16x4_f32 into LDS,
//   top-20 per row, edge softmax w/ self loop, GAT aggregate -> gnn_pre
// ---------------------------------------------------------------------------
#define SC_STRIDE 1028  // 1024 + 4 pad: de-conflicts LDS banks between halves

__global__ __launch_bounds__(256) void k_scores_gat(
    const float* __restrict__ mixed, const float* __restrict__ xlin,
    const float* __restrict__ s_i, const float* __restrict__ s_j,
    const float* __restrict__ b_gnn, float* __restrict__ gnn_pre) {
  __shared__ float sc[16 * SC_STRIDE];   // 65792 B score strip
  __shared__ int tIdx[16][Kk];
  __shared__ float tVal[16][Kk];
  __shared__ float tW[16][Kk + 1];       // fused att*topk_w, [Kk] = self weight

  const int b = blockIdx.x >> 6;         // 64 strips per batch
  const int i0 = (blockIdx.x & 63) * 16;
  const int tid = threadIdx.x;
  const int wave = tid >> 5;
  const int lane = tid & 31;
  const int half = lane >> 4;            // wave32 halves
  const int l16 = lane & 15;

  const float* mb = mixed + (size_t)b * Nn * Dd;

  // A fragment (16x4 f32 per step): lanes0-15 K={0,1}, lanes16-31 K={2,3}
  v2f A[16];
  {
    const float* ar = mb + (size_t)(i0 + l16) * Dd + 2 * half;
#pragma unroll
    for (int t = 0; t < 16; ++t) A[t] = *(const v2f*)(ar + 4 * t);
  }

  for (int jt = 0; jt < 8; ++jt) {
    const int j0 = (jt * 8 + wave) * 16;
    v2f Bv[16];
    const float* br = mb + (size_t)(j0 + l16) * Dd + 2 * half;
#pragma unroll
    for (int t = 0; t < 16; ++t) Bv[t] = *(const v2f*)(br + 4 * t);
    v8f acc = {};
#pragma unroll
    for (int t = 0; t < 16; ++t)
      acc = __builtin_amdgcn_wmma_f32_16x16x4_f32(false, A[t], false, Bv[t],
                                                  (short)0, acc, false, false);
    // D layout: row = v + 8*half, col = l16
#pragma unroll
    for (int v = 0; v < 8; ++v)
      sc[(v + 8 * half) * SC_STRIDE + j0 + l16] = nanfix(acc[v]);
  }
  __syncthreads();

  // ---- top-20 per row; wave w owns rows 2w (lanes 0-15) / 2w+1 (lanes 16-31)
  const int row = wave * 2 + half;
  const int gi = i0 + row;
  float* srow = &sc[row * SC_STRIDE];
  for (int k = 0; k < Kk; ++k) {
    float bv = -INFINITY;
    int bi = -1;
    for (int c = l16; c < Nn; c += 16) {
      float v = srow[c];
      bool gt = v > bv;
      bv = gt ? v : bv;
      bi = gt ? c : bi;
    }
#pragma unroll
    for (int off = 8; off; off >>= 1) {
      float ov = __shfl_xor(bv, off, 16);
      int oi = __shfl_xor(bi, off, 16);
      bool take = (ov > bv) || (ov == bv && oi < bi);
      bv = take ? ov : bv;
      bi = take ? oi : bi;
    }
    if (l16 == 0) {
      tIdx[row][k] = bi;
      tVal[row][k] = bv;
      srow[bi] = -INFINITY;  // extract
    }
    asm volatile("s_wait_dscnt 0" ::: "memory");  // LDS RAW, same-wave
  }

  // ---- per-row softmaxes (topk_w and GAT attention), serial on group leader
  if (l16 == 0) {
    const float si = s_i[(size_t)b * Nn + gi];
    const float sji = s_j[(size_t)b * Nn + gi];
    // softmax over topk values
    float m1 = tVal[row][0];
#pragma unroll
    for (int k = 1; k < Kk; ++k) m1 = fmaxf(m1, tVal[row][k]);
    float tw[Kk], s1 = 0.f;
#pragma unroll
    for (int k = 0; k < Kk; ++k) {
      tw[k] = expf(tVal[row][k] - m1);
      s1 += tw[k];
    }
    // attention logits (self edges in topk masked -inf) + appended self loop
    float al[Kk + 1];
#pragma unroll
    for (int k = 0; k < Kk; ++k) {
      int id = tIdx[row][k];
      float a = lrelu(si + s_j[(size_t)b * Nn + id]);
      al[k] = (id == gi) ? -INFINITY : a;
    }
    al[Kk] = lrelu(si + sji);
    float m2 = al[Kk];
#pragma unroll
    for (int k = 0; k < Kk; ++k) m2 = fmaxf(m2, al[k]);
    float s2 = 0.f, ea[Kk + 1];
#pragma unroll
    for (int k = 0; k <= Kk; ++k) {
      ea[k] = expf(al[k] - m2);
      s2 += ea[k];
    }
#pragma unroll
    for (int k = 0; k < Kk; ++k) tW[row][k] = (ea[k] / s2) * (tw[k] / s1);
    tW[row][Kk] = ea[Kk] / s2;
  }
  asm volatile("s_wait_dscnt 0" ::: "memory");

  // ---- aggregate: lane l16 owns contiguous d slice [4*l16, 4*l16+4):
  //      21 coalesced global_load_b128 gathers + 1 global_store_b128
  const float* xb = xlin + (size_t)b * Nn * Dd;
  const int d0 = 4 * l16;
  const float ws = tW[row][Kk];
  v4f accv = *(const v4f*)(b_gnn + d0);
  {
    v4f xs = *(const v4f*)(xb + (size_t)gi * Dd + d0);
    accv.x += ws * xs.x; accv.y += ws * xs.y;
    accv.z += ws * xs.z; accv.w += ws * xs.w;
  }
  for (int k = 0; k < Kk; ++k) {
    float wk = tW[row][k];
    v4f xs = *(const v4f*)(xb + (size_t)tIdx[row][k] * Dd + d0);
    accv.x += wk * xs.x; accv.y += wk * xs.y;
    accv.z += wk * xs.z; accv.w += wk * xs.w;
  }
  *(v4f*)(gnn_pre + ((size_t)b * Nn + gi) * Dd + d0) = accv;
}

// ---------------------------------------------------------------------------
// K4: per-channel sum / sumsq over all 32768 rows (atomicAdd into acc[0:128])
// ---------------------------------------------------------------------------
__global__ __launch_bounds__(256) void k_bn_stats(const float* __restrict__ src,
                                                  float* __restrict__ acc) {
  __shared__ float sh[256], sh2[256];
  const int tid = threadIdx.x;
  const int d = tid & 63;
  const int sub = tid >> 6;  // 0..3
  float s = 0.f, s2 = 0.f;
  for (size_t r = (size_t)blockIdx.x * 4 + sub; r < BNTOT; r += 1024) {
    float v = src[r * Dd + d];
    s += v;
    s2 += v * v;
  }
  sh[tid] = s;
  sh2[tid] = s2;
  __syncthreads();
  if (sub == 0) {
#pragma unroll
    for (int k = 1; k < 4; ++k) {
      s += sh[tid + 64 * k];
      s2 += sh2[tid + 64 * k];
    }
    atomicAdd(&acc[d], s);
    atomicAdd(&acc[64 + d], s2);
  }
}

// ---------------------------------------------------------------------------
// K5: out2 = relu(BN1(gnn_pre)) * node_embed
// ---------------------------------------------------------------------------
__global__ __launch_bounds__(256) void k_bn1_apply(
    const float* __restrict__ gnn_pre, const float* __restrict__ acc1,
    const float* __restrict__ gamma, const float* __restrict__ beta,
    const float* __restrict__ embed, float* __restrict__ out2) {
  const int idx = blockIdx.x * 256 + threadIdx.x;
  const int n = idx & (Nn - 1);
  const float inv = 1.f / (float)BNTOT;
#pragma unroll 8
  for (int d = 0; d < Dd; ++d) {
    float mu = acc1[d] * inv;
    float var = acc1[64 + d] * inv - mu * mu;
    float g = (gnn_pre[(size_t)idx * Dd + d] - mu) * rsqrtf(var + 1e-5f) * gamma[d] + beta[d];
    g = fmaxf(g, 0.f);
    out2[(size_t)idx * Dd + d] = g * embed[(size_t)n * Dd + d];
  }
}

// ---------------------------------------------------------------------------
// K6: out = relu(BN2(out2)) @ W_out + b_out
// ---------------------------------------------------------------------------
__global__ __launch_bounds__(256) void k_bn2_out(
    const float* __restrict__ out2, const float* __restrict__ acc2,
    const float* __restrict__ gamma, const float* __restrict__ beta,
    const float* __restrict__ W_out, const float* __restrict__ b_out,
    float* __restrict__ out) {
  const int idx = blockIdx.x * 256 + threadIdx.x;
  const float inv = 1.f / (float)BNTOT;
  float accv = b_out[0];
#pragma unroll 8
  for (int d = 0; d < Dd; ++d) {
    float mu = acc2[d] * inv;
    float var = acc2[64 + d] * inv - mu * mu;
    float g = (out2[(size_t)idx * Dd + d] - mu) * rsqrtf(var + 1e-5f) * gamma[d] + beta[d];
    accv += fmaxf(g, 0.f) * W_out[d];
  }
  out[idx] = accv;
}

// ---------------------------------------------------------------------------
extern "C" void kernel_launch(void* const* d_in, const int* in_sizes, int n_in,
                              void* d_out, int out_size, void* d_ws, size_t ws_size,
                              hipStream_t stream) {
  const float* data = (const float*)d_in[0];
  const float* gumbel = (const float*)d_in[1];
  const float* W_cond = (const float*)d_in[2];
  const float* b_cond = (const float*)d_in[3];
  const float* W_ap = (const float*)d_in[4];
  const float* b_ap = (const float*)d_in[5];
  const float* W_av = (const float*)d_in[6];
  const float* W_r = (const float*)d_in[7];
  const float* b_r = (const float*)d_in[8];
  const float* e_base = (const float*)d_in[9];
  const float* lr_u = (const float*)d_in[10];
  const float* lr_v = (const float*)d_in[11];
  const float* W_gnn = (const float*)d_in[12];
  const float* att_i = (const float*)d_in[13];
  const float* att_j = (const float*)d_in[14];
  const float* att_em_i = (const float*)d_in[15];
  const float* att_em_j = (const float*)d_in[16];
  const float* b_gnn = (const float*)d_in[17];
  const float* bn1_g = (const float*)d_in[18];
  const float* bn1_b = (const float*)d_in[19];
  const float* embed = (const float*)d_in[20];
  const float* bn2_g = (const float*)d_in[21];
  const float* bn2_b = (const float*)d_in[22];
  const float* W_out = (const float*)d_in[23];
  const float* b_out = (const float*)d_in[24];

  float* out = (float*)d_out;               // [B*N]
  float* out_hsys = out + BNTOT;            // [B*D]
  float* out_pisoft = out_hsys + Bt * Dd;   // [B*M]

  float* w = (float*)d_ws;
  float* bufA = w;                // h_it, later gnn_pre
  float* bufB = bufA + BIGSZ;     // x_lin
  float* bufC = bufB + BIGSZ;     // mixed, later out2
  float* e_it = bufC + BIGSZ;     // [32768]
  float* s_i = e_it + BNTOT;
  float* s_j = s_i + BNTOT;
  float* pi_t = s_j + BNTOT;      // [128]
  float* acc1 = pi_t + 128;       // [128] sum|sumsq
  float* acc2 = acc1 + 128;       // [128]

  k_node_prep<<<BNTOT / 256, 256, 0, stream>>>(data, W_cond, b_cond, W_ap, b_ap,
                                               W_av, W_gnn, att_i, att_j, bufA,
                                               e_it, bufB, s_i, s_j);
  k_pool_route<<<Bt, 256, 0, stream>>>(bufA, e_it, W_r, b_r, gumbel, out_hsys,
                                       out_pisoft, pi_t);
  k_mixed<<<BNTOT / 256, 256, 0, stream>>>(pi_t, e_base, lr_u, lr_v, att_em_i,
                                           att_em_j, bufC, s_i, s_j);
  k_scores_gat<<<Bt * (Nn / 16), 256, 0, stream>>>(bufC, bufB, s_i, s_j, b_gnn,
                                                   bufA);  // gnn_pre -> bufA
  hipMemsetAsync(acc1, 0, 256 * sizeof(float), stream);    // zero acc1+acc2
  k_bn_stats<<<256, 256, 0, stream>>>(bufA, acc1);
  k_bn1_apply<<<BNTOT / 256, 256, 0, stream>>>(bufA, acc1, bn1_g, bn1_b, embed,
                                               bufC);      // out2 -> bufC
  k_bn_stats<<<256, 256, 0, stream>>>(bufC, acc2);
  k_bn2_out<<<BNTOT / 256, 256, 0, stream>>>(bufC, acc2, bn2_g, bn2_b, W_out,
                                             b_out, out);
  (void)in_sizes; (void)n_in; (void)out_size; (void)ws_size;
}